// HAFIS_Module_6889127543034
// MI455X (gfx1250) — compile-verified
//
#include <hip/hip_runtime.h>
#include <hip/hip_bf16.h>
#include <math.h>

typedef __attribute__((ext_vector_type(2))) float v2f;
typedef __attribute__((ext_vector_type(8))) float v8f;

#define CB   256
#define HWB  (128 * 128)
#define EPSQ 1e-6f

__device__ __forceinline__ float sigmoidf(float x) {
    return 1.0f / (1.0f + expf(-x));
}

__global__ __launch_bounds__(256) void hafis_fused_kernel(
    const float* __restrict__ features,
    const float* __restrict__ w1, const float* __restrict__ b1,   // conv1: (16,3),(16)
    const float* __restrict__ w2, const float* __restrict__ b2,   // conv2: (1,16),(1)
    const float* __restrict__ l1w, const float* __restrict__ l1b, // lin1: (32,3),(32)
    const float* __restrict__ l2w, const float* __restrict__ l2b, // lin2: (1,32),(1)
    const int*   __restrict__ snr_p,
    float* __restrict__ out_q, float* __restrict__ out_imp, float* __restrict__ out_bits)
{
    const int p  = blockIdx.x * blockDim.x + threadIdx.x;  // pixel id in [0, B*H*W)
    const int b  = p >> 14;                                // / (H*W)
    const int hw = p & (HWB - 1);
    const float* fp = features + (size_t)b * CB * HWB + hw;

    // ---------------- Pass 1: channel statistics (streaming, coalesced) ----
    float sum = 0.f, sumsq = 0.f, sumabs = 0.f;
    float fmn = 3.4e38f, fmx = -3.4e38f;
    #pragma unroll 8
    for (int c = 0; c < CB; ++c) {
        if ((c & 7) == 0)  // one speculative prefetch per unrolled body, 32 ch ahead
            __builtin_prefetch(fp + (size_t)(c + 32) * HWB, 0, 0);
        float f = fp[(size_t)c * HWB];
        sum    += f;
        sumsq   = fmaf(f, f, sumsq);
        sumabs += fabsf(f);
        fmn = fminf(fmn, f);
        fmx = fmaxf(fmx, f);
    }

    const float inv_C   = 1.0f / 256.0f;
    float mean     = sum * inv_C;
    float mean_sq  = sumsq * inv_C;
    float mean_abs = sumabs * inv_C;
    float var_raw  = fmaxf(sumsq - sum * mean, 0.0f) * (1.0f / 255.0f); // ddof=1
    float stdv     = sqrtf(var_raw);
    float mag  = sqrtf(mean_sq);                 // sqrt(sum^2)/sqrt(C)
    float varn = var_raw / (mean_sq + EPSQ);
    float stdn = stdv    / (mean_abs + EPSQ);
    mag  = fminf(fmaxf(mag,  0.f), 1.f);
    varn = fminf(fmaxf(varn, 0.f), 1.f);
    stdn = fminf(fmaxf(stdn, 0.f), 1.f);

    // ---------------- conv1 (3->16) as exact f32 WMMA 16x16x4 -------------
    // Wave32 = 32 pixels = two 16-pixel tiles.  A = W1 (16x4, K3 zero-padded),
    // B = fuzzy (4x16 per tile), C = bias broadcast.  f32 A 16x4 layout:
    //   VGPR0: lanes0-15 K=0, lanes16-31 K=2 ; VGPR1: lanes0-15 K=1, lanes16-31 K=3
    const int  lane = threadIdx.x & 31;
    const bool hi   = lane >= 16;
    const int  m    = lane & 15;

    v2f A;
    A.x = hi ? w1[m * 3 + 2] : w1[m * 3 + 0];
    A.y = hi ? 0.0f          : w1[m * 3 + 1];

    v8f Cm;
    #pragma unroll
    for (int r = 0; r < 8; ++r) Cm[r] = b1[(hi ? 8 : 0) + r];

    // B tile0: columns = pixels of lanes 0..15 (K0=mag,K1=var,K2=std,K3=0)
    float std_lo = __shfl(stdn, m, 32);            // lanes16+L get std of lane L
    v2f B0; B0.x = hi ? std_lo : mag;  B0.y = hi ? 0.0f : varn;
    // B tile1: columns = pixels of lanes 16..31
    float mag_hi = __shfl(mag,  lane | 16, 32);
    float var_hi = __shfl(varn, lane | 16, 32);
    v2f B1; B1.x = hi ? stdn : mag_hi; B1.y = hi ? 0.0f : var_hi;

    v8f D0 = __builtin_amdgcn_wmma_f32_16x16x4_f32(false, A, false, B0, (short)0, Cm, false, false);
    v8f D1 = __builtin_amdgcn_wmma_f32_16x16x4_f32(false, A, false, B1, (short)0, Cm, false, false);

    // conv2 (16->1): each lane holds 8 hidden rows of one column; relu + dot,
    // then pair-sum across the lane^16 partner.
    float p0 = 0.f, p1 = 0.f;
    #pragma unroll
    for (int r = 0; r < 8; ++r) {
        float w = w2[(hi ? 8 : 0) + r];
        p0 = fmaf(w, fmaxf(D0[r], 0.f), p0);
        p1 = fmaf(w, fmaxf(D1[r], 0.f), p1);
    }
    p0 += __shfl_xor(p0, 16, 32);
    p1 += __shfl_xor(p1, 16, 32);
    float bias2 = b2[0];
    float imp = sigmoidf((hi ? p1 : p0) + bias2);   // my own pixel's importance

    // ---------------- bits MLP 3->32->1 (snr & rate are uniform) ----------
    float snr_norm = (float)snr_p[0] * (1.0f / 30.0f);
    float acc = l2b[0];
    #pragma unroll 8
    for (int o = 0; o < 32; ++o) {
        float z = fmaf(l1w[o * 3 + 0], imp,
                  fmaf(l1w[o * 3 + 1], snr_norm,
                  fmaf(l1w[o * 3 + 2], 0.5f, l1b[o])));
        acc = fmaf(l2w[o], fmaxf(z, 0.f), acc);
    }
    float bits = fmaf(8.0f, sigmoidf(acc), 4.0f);   // MIN_BITS + b*(MAX-MIN)

    // ---------------- Pass 2: quantize (reads hit L2: 256KB/block set) ----
    float levels  = exp2f(bits);
    float lm1     = levels - 1.0f;
    float inv_lm1 = 1.0f / lm1;
    float denom   = (fmx - fmn) + EPSQ;
    float inv_d   = 1.0f / denom;
    float* qp = out_q + (size_t)b * CB * HWB + hw;
    #pragma unroll 8
    for (int c = 0; c < CB; ++c) {
        float f  = fp[(size_t)c * HWB];
        float fn = (f - fmn) * inv_d;
        float q  = rintf(fn * lm1) * inv_lm1;       // round-half-even == jnp.round
        qp[(size_t)c * HWB] = fmaf(q, denom, fmn);
    }

    out_imp[p]  = imp;
    out_bits[p] = bits;
}

extern "C" void kernel_launch(void* const* d_in, const int* in_sizes, int n_in,
                              void* d_out, int out_size, void* d_ws, size_t ws_size,
                              hipStream_t stream) {
    (void)in_sizes; (void)n_in; (void)out_size; (void)d_ws; (void)ws_size;
    const float* features = (const float*)d_in[0];
    const float* w1  = (const float*)d_in[1];
    const float* b1  = (const float*)d_in[2];
    const float* w2  = (const float*)d_in[3];
    const float* b2  = (const float*)d_in[4];
    const float* l1w = (const float*)d_in[5];
    const float* l1b = (const float*)d_in[6];
    const float* l2w = (const float*)d_in[7];
    const float* l2b = (const float*)d_in[8];
    const int*   snr = (const int*)d_in[9];

    const size_t NTOT = (size_t)16 * 256 * 128 * 128;  // quantized elements
    const size_t NPIX = (size_t)16 * 128 * 128;        // importance / bits

    float* out      = (float*)d_out;
    float* out_q    = out;
    float* out_imp  = out + NTOT;
    float* out_bits = out + NTOT + NPIX;

    dim3 block(256);
    dim3 grid((unsigned)(NPIX / 256));  // 1024 blocks, exact cover, no guards
    hafis_fused_kernel<<<grid, block, 0, stream>>>(
        features, w1, b1, w2, b2, l1w, l1b, l2w, l2b, snr,
        out_q, out_imp, out_bits);
}